// OPT_mlp_pe_67405216744072
// MI455X (gfx1250) — compile-verified
//
#include <hip/hip_runtime.h>

typedef float v2f __attribute__((ext_vector_type(2)));
typedef float v4f __attribute__((ext_vector_type(4)));
typedef float v8f __attribute__((ext_vector_type(8)));

// ---------------------------------------------------------------------------
// Build A = I - S and X0 = I + S,  S = 0.5*(R - R^T), batched over r blocks.
// ---------------------------------------------------------------------------
__global__ __launch_bounds__(256) void OPT_cayley_init(
    const float* __restrict__ R, float* __restrict__ A, float* __restrict__ X,
    int blk) {
  long idx = (long)blockIdx.x * 256 + threadIdx.x;
  long per = (long)blk * blk;
  int r = (int)(idx / per);
  int rem = (int)(idx - (long)r * per);
  int i = rem / blk;
  int j = rem - i * blk;
  long base = (long)r * per;
  float s = 0.5f * (R[base + (long)i * blk + j] - R[base + (long)j * blk + i]);
  float id = (i == j) ? 1.0f : 0.0f;
  A[idx] = id - s;
  X[idx] = id + s;
}

// ---------------------------------------------------------------------------
// Tiled fp32 WMMA GEMM.  C = P @ B  (optionally B transposed), epilogues:
//   MODE 0 : C = P@B
//   MODE 1 : C = 2*D - P@B          (Newton-Schulz step / final Cayley)
//   MODE 2 : C = P@B + bias[n]      (final linear layer)
//
// Tile: BM=BN=128, BK=32, 256 threads = 8 waves (2x4), wave tile 64x32.
// Both LDS tiles are stored outer-dim-major with padded stride 36 so every
// fragment (A and B) is one aligned ds_load_b64 (k = 2*laneHi + vgpr matches
// the V_WMMA_F32_16X16X4_F32 operand layouts, ISA 7.12.2).  Global->LDS is
// software-pipelined through registers to overlap HBM latency with WMMA.
//
// Cache policy (192 MB L2 plan): in the dominant NT+bias GEMM, W (256 MB) is
// read once -> non-temporal loads; out (512 MB) is written once -> non-
// temporal stores.  This keeps the 128 MB y panel L2-resident across the 128
// N-tiles that re-read it.  All dims are multiples of the tile sizes.
// ---------------------------------------------------------------------------
template <int MODE, bool TRANSB>
__global__ __launch_bounds__(256) void OPT_gemm_f32_wmma(
    const float* __restrict__ P, const float* __restrict__ B,
    const float* __restrict__ D, const float* __restrict__ bias,
    float* __restrict__ C, int K, long lda, long ldb, long ldc, long ldd,
    long sP, long sB, long sC, long sD) {
  constexpr int BM = 128, BN = 128, BK = 32;
  constexpr int TSTR = BK + 4;  // padded LDS stride (bank-conflict free reads)
  __shared__ float As[BM * TSTR];
  __shared__ float Bs[BN * TSTR];

  const int n0 = blockIdx.x * BN;
  const int m0 = blockIdx.y * BM;
  const int bat = blockIdx.z;

  const float* Pb = P + sP * bat;
  const float* Bb = B + sB * bat;
  const float* Db = D + sD * bat;
  float* Cb = C + sC * bat;

  const int t = threadIdx.x;
  const int wave = t >> 5;
  const int lane = t & 31;
  const int lr = lane & 15;   // m (A) / n (B,C) within 16
  const int lh = lane >> 4;   // half-wave select
  const int wr = wave >> 2;   // 0..1 -> 64-row slab
  const int wc = wave & 3;    // 0..3 -> 32-col slab
  const int wave_m = wr * 64;
  const int wave_n = wc * 32;

  // Per-lane LDS fragment base offsets (k = 2*lh + vgpr within a float2)
  int aoff[4];
#pragma unroll
  for (int a = 0; a < 4; ++a)
    aoff[a] = (wave_m + a * 16 + lr) * TSTR + lh * 2;
  int boff[2];
#pragma unroll
  for (int b = 0; b < 2; ++b)
    boff[b] = (wave_n + b * 16 + lr) * TSTR + lh * 2;

  // Global staging coordinates (each thread moves 16 floats per tile & side)
  const int am = t >> 1;           // A row 0..127
  const int akb = (t & 1) * 16;    // A col base {0,16}
  const int tn = t >> 1;           // B col (NT) 0..127
  const int tkb = (t & 1) * 16;    // B row base (NT) {0,16}
  const int nk = t >> 3;           // B row (NN) 0..31
  const int nnb = (t & 7) * 16;    // B col base (NN)

  v8f acc[4][2];
#pragma unroll
  for (int a = 0; a < 4; ++a)
#pragma unroll
    for (int b = 0; b < 2; ++b) acc[a][b] = (v8f)0.0f;

  v4f pa[4], pb[4];
  const float* gA = Pb + (long)(m0 + am) * lda + akb;
  const float* gB = TRANSB ? (Bb + (long)(n0 + tn) * ldb + tkb)
                           : (Bb + (long)nk * ldb + n0 + nnb);

  // prime the pipeline: tile k0 = 0 into registers
#pragma unroll
  for (int i = 0; i < 4; ++i) pa[i] = *(const v4f*)(gA + 4 * i);
#pragma unroll
  for (int i = 0; i < 4; ++i) {
    if (TRANSB)  // W streams exactly once: do not pollute L2
      pb[i] = __builtin_nontemporal_load((const v4f*)(gB + 4 * i));
    else
      pb[i] = *(const v4f*)(gB + 4 * i);
  }

  for (int k0 = 0;;) {
    // ---- registers -> LDS ----
#pragma unroll
    for (int i = 0; i < 4; ++i)
      *(v4f*)&As[am * TSTR + akb + 4 * i] = pa[i];
    if (TRANSB) {
#pragma unroll
      for (int i = 0; i < 4; ++i)
        *(v4f*)&Bs[tn * TSTR + tkb + 4 * i] = pb[i];
    } else {
#pragma unroll
      for (int i = 0; i < 4; ++i) {
        Bs[(nnb + 4 * i + 0) * TSTR + nk] = pb[i].x;
        Bs[(nnb + 4 * i + 1) * TSTR + nk] = pb[i].y;
        Bs[(nnb + 4 * i + 2) * TSTR + nk] = pb[i].z;
        Bs[(nnb + 4 * i + 3) * TSTR + nk] = pb[i].w;
      }
    }
    __syncthreads();

    // ---- prefetch next tile into registers (overlaps with WMMA below) ----
    const int k1 = k0 + BK;
    const bool more = (k1 < K);
    if (more) {
      const float* nA = gA + k1;
      const float* nB = TRANSB ? (gB + k1) : (gB + (long)k1 * ldb);
#pragma unroll
      for (int i = 0; i < 4; ++i) pa[i] = *(const v4f*)(nA + 4 * i);
#pragma unroll
      for (int i = 0; i < 4; ++i) {
        if (TRANSB)
          pb[i] = __builtin_nontemporal_load((const v4f*)(nB + 4 * i));
        else
          pb[i] = *(const v4f*)(nB + 4 * i);
      }
    }

    // ---- 8 k-steps x 8 WMMA from LDS ----
#pragma unroll
    for (int ks = 0; ks < BK / 4; ++ks) {
      v2f af[4];
#pragma unroll
      for (int a = 0; a < 4; ++a)
        af[a] = *(const v2f*)&As[aoff[a] + ks * 4];
      v2f bf[2];
#pragma unroll
      for (int b = 0; b < 2; ++b)
        bf[b] = *(const v2f*)&Bs[boff[b] + ks * 4];
#pragma unroll
      for (int a = 0; a < 4; ++a)
#pragma unroll
        for (int b = 0; b < 2; ++b)
          acc[a][b] = __builtin_amdgcn_wmma_f32_16x16x4_f32(
              false, af[a], false, bf[b], (short)0, acc[a][b], false, false);
    }
    __syncthreads();
    if (!more) break;
    k0 = k1;
  }

  // ---- epilogue ----
  float bv[2] = {0.0f, 0.0f};
  if (MODE == 2) {
#pragma unroll
    for (int b = 0; b < 2; ++b) bv[b] = bias[n0 + wave_n + b * 16 + lr];
  }
#pragma unroll
  for (int a = 0; a < 4; ++a) {
#pragma unroll
    for (int b = 0; b < 2; ++b) {
      long mbase = (long)m0 + wave_m + a * 16 + lh * 8;
      long nn = (long)n0 + wave_n + b * 16 + lr;
#pragma unroll
      for (int e = 0; e < 8; ++e) {
        long m = mbase + e;
        float v = acc[a][b][e];
        if (MODE == 0) {
          Cb[m * ldc + nn] = v;
        } else if (MODE == 1) {
          Cb[m * ldc + nn] = 2.0f * Db[m * ldd + nn] - v;
        } else {
          // final 512 MB output: written once, never re-read -> stream past L2
          __builtin_nontemporal_store(v + bv[b], &Cb[m * ldc + nn]);
        }
      }
    }
  }
}

// ---------------------------------------------------------------------------
// Host launch:  x[4,2048,4096], W[16384,4096], bias[16384], R[4,1024,1024]
//   1) A = I-S, X = I+S
//   2) 8x Newton-Schulz: T = A@X ; X' = 2X - X@T   (residual -> S^512 ~ 0)
//   3) Q = 2X - A@X
//   4) y = x @ blockdiag(Q)     (batched NN, strided)
//   5) out = y @ W^T + bias     (NT + bias, memory-bound ~0.9 GB)
// ---------------------------------------------------------------------------
extern "C" void kernel_launch(void* const* d_in, const int* in_sizes, int n_in,
                              void* d_out, int out_size, void* d_ws,
                              size_t ws_size, hipStream_t stream) {
  const float* x = (const float*)d_in[0];
  const float* W = (const float*)d_in[1];
  const float* bias = (const float*)d_in[2];
  const float* R = (const float*)d_in[3];
  float* out = (float*)d_out;

  const int r = 4, blk = 1024;
  const int in_f = r * blk;                 // 4096
  const int out_f = in_sizes[2];            // 16384
  const int M = in_sizes[0] / in_f;         // 8192 (= 4*2048 rows)

  const long NB = (long)r * blk * blk;      // 4M floats per batched square buf
  float* w = (float*)d_ws;
  float* Am = w;
  float* Xp = w + NB;
  float* Tp = w + 2 * NB;
  float* X2 = w + 3 * NB;
  float* Yp = w + 4 * NB;                   // M x in_f

  // 1) init A, X
  OPT_cayley_init<<<(unsigned)(NB / 256), 256, 0, stream>>>(R, Am, Xp, blk);

  const long S = (long)blk * blk;
  dim3 gSq(blk / 128, blk / 128, r);
  // 2) Newton-Schulz iterations
  for (int it = 0; it < 8; ++it) {
    OPT_gemm_f32_wmma<0, false><<<gSq, 256, 0, stream>>>(
        Am, Xp, nullptr, nullptr, Tp, blk, blk, blk, blk, 0, S, S, S, 0);
    OPT_gemm_f32_wmma<1, false><<<gSq, 256, 0, stream>>>(
        Xp, Tp, Xp, nullptr, X2, blk, blk, blk, blk, blk, S, S, S, S);
    float* tmp = Xp; Xp = X2; X2 = tmp;
  }
  // 3) Q = 2X - A@X   (stored in X2)
  OPT_gemm_f32_wmma<1, false><<<gSq, 256, 0, stream>>>(
      Am, Xp, Xp, nullptr, X2, blk, blk, blk, blk, blk, S, S, S, S);
  float* Qp = X2;

  // 4) y = x @ blockdiag(Q): batched over r, A/C column-offset by r*blk
  dim3 gY(blk / 128, M / 128, r);
  OPT_gemm_f32_wmma<0, false><<<gY, 256, 0, stream>>>(
      x, Qp, nullptr, nullptr, Yp, blk, in_f, blk, in_f, 0,
      (long)blk, S, (long)blk, 0);

  // 5) out = y @ W^T + bias
  dim3 gO(out_f / 128, M / 128, 1);
  OPT_gemm_f32_wmma<2, true><<<gO, 256, 0, stream>>>(
      Yp, W, nullptr, bias, out, in_f, in_f, in_f, out_f, 0, 0, 0, 0, 0);
}